// SAGEConv_37151467110629
// MI455X (gfx1250) — compile-verified
//
#include <hip/hip_runtime.h>

// SAGEConv: out = (segment_mean(x[src] -> dst)) @ W_l + b_l + x @ W_r + b_r
// Phase 1: scatter-mean (bandwidth/atomic bound, dominates).
// Phase 2: fused dual GEMM on the fp32 WMMA pipe (v_wmma_f32_16x16x4_f32).

typedef __attribute__((ext_vector_type(2))) float v2f;
typedef __attribute__((ext_vector_type(8))) float v8f;

#define FEAT 64

__global__ void zero_ws_kernel(float* __restrict__ p, int n) {
    int i = blockIdx.x * blockDim.x + threadIdx.x;
    int stride = gridDim.x * blockDim.x;
    for (; i < n; i += stride) p[i] = 0.0f;
}

// One wave32 per edge: 32 lanes x float2 = 64 features.
__global__ void scatter_edges_kernel(const float* __restrict__ x,
                                     const int* __restrict__ ei,
                                     float* __restrict__ agg,
                                     float* __restrict__ cnt,
                                     int n_edges) {
    long long idx = (long long)blockIdx.x * blockDim.x + threadIdx.x;
    int e = (int)(idx >> 5);
    if (e >= n_edges) return;
    int lane = (int)(idx & 31);
    int src = ei[e];              // edge_index[0][e]
    int dst = ei[n_edges + e];    // edge_index[1][e]

    const float2 v = *(const float2*)(x + (long long)src * FEAT + lane * 2);
    float* a = agg + (long long)dst * FEAT + lane * 2;
    atomicAdd(a + 0, v.x);        // global_atomic_add_f32
    atomicAdd(a + 1, v.y);
    if (lane == 0) atomicAdd(cnt + dst, 1.0f);
}

// One wave per 16x16 output tile. K=64 consumed 4 at a time with
// V_WMMA_F32_16X16X4_F32 (exact fp32 tensor path, CDNA5-only shape).
// A layout (16x4 f32): lane L holds A[L&15][2*(L>>4) + {0,1}] in {v0,v1}.
// B layout (4x16 f32): lane L holds B[2*(L>>4) + {0,1}][L&15] in {v0,v1}.
// C/D layout: VGPR r, lane L -> row r + 8*(L>>4), col L&15.
__global__ void __launch_bounds__(128) sage_gemm_kernel(
        const float* __restrict__ x,
        const float* __restrict__ agg,
        const float* __restrict__ cnt,
        const float* __restrict__ Wl, const float* __restrict__ bl,
        const float* __restrict__ Wr, const float* __restrict__ br,
        float* __restrict__ out, int n_nodes) {
    const int lane = threadIdx.x & 31;
    const int wave = threadIdx.x >> 5;     // 0..3 -> output column tile
    const int row0 = blockIdx.x * 16;
    const int rc   = lane & 15;            // A row in tile == C column in tile
    const int h    = lane >> 4;            // half-wave selector
    const int ncol = wave * 16 + rc;       // global output column (D=64)

    int arow = row0 + rc;
    if (arow >= n_nodes) arow = n_nodes - 1;   // clamp (select, not EXEC mask)

    const float inv = 1.0f / fmaxf(cnt[arow], 1.0f);
    const float* aggRow = agg + (long long)arow * FEAT;
    const float* xRow   = x   + (long long)arow * FEAT;

    v8f acc = {};
#pragma unroll
    for (int kk = 0; kk < 16; ++kk) {
        const int k = kk * 4 + 2 * h;      // even -> 8B-aligned float2 loads

        // mean tile @ W_l
        float2 am = *(const float2*)(aggRow + k);
        v2f a;  a.x  = am.x * inv;         a.y  = am.y * inv;
        v2f b;  b.x  = Wl[k * FEAT + ncol]; b.y = Wl[(k + 1) * FEAT + ncol];
        acc = __builtin_amdgcn_wmma_f32_16x16x4_f32(
                  false, a, false, b, (short)0, acc, false, false);

        // x tile @ W_r
        float2 ax = *(const float2*)(xRow + k);
        v2f a2; a2.x = ax.x;               a2.y = ax.y;
        v2f b2; b2.x = Wr[k * FEAT + ncol]; b2.y = Wr[(k + 1) * FEAT + ncol];
        acc = __builtin_amdgcn_wmma_f32_16x16x4_f32(
                  false, a2, false, b2, (short)0, acc, false, false);
    }

    const float bias = bl[ncol] + br[ncol];
#pragma unroll
    for (int r = 0; r < 8; ++r) {
        int orow = row0 + r + 8 * h;
        if (orow < n_nodes)
            out[(long long)orow * FEAT + ncol] = acc[r] + bias;
    }
}

extern "C" void kernel_launch(void* const* d_in, const int* in_sizes, int n_in,
                              void* d_out, int out_size, void* d_ws, size_t ws_size,
                              hipStream_t stream) {
    const float* x  = (const float*)d_in[0];
    const int*   ei = (const int*)d_in[1];
    const float* Wl = (const float*)d_in[2];
    const float* bl = (const float*)d_in[3];
    const float* Wr = (const float*)d_in[4];
    const float* br = (const float*)d_in[5];
    float* out = (float*)d_out;

    const int n_nodes = in_sizes[0] / FEAT;
    const int n_edges = in_sizes[1] / 2;

    // Workspace layout: agg[n_nodes*64] then cnt[n_nodes]  (~26 MB)
    float* agg = (float*)d_ws;
    float* cnt = agg + (size_t)n_nodes * FEAT;

    // 1) zero agg + cnt (contiguous)
    const int zn = n_nodes * (FEAT + 1);
    zero_ws_kernel<<<2048, 256, 0, stream>>>(agg, zn);

    // 2) scatter: one wave per edge
    const long long sthreads = (long long)n_edges * 32;
    const int sblocks = (int)((sthreads + 255) / 256);
    scatter_edges_kernel<<<sblocks, 256, 0, stream>>>(x, ei, agg, cnt, n_edges);

    // 3) fused dual GEMM + bias on the fp32 WMMA pipe
    const int gblocks = (n_nodes + 15) / 16;
    sage_gemm_kernel<<<gblocks, 128, 0, stream>>>(x, agg, cnt, Wl, bl, Wr, br,
                                                  out, n_nodes);
}